// EdgeControl_61658550502079
// MI455X (gfx1250) — compile-verified
//
#include <hip/hip_runtime.h>
#include <math.h>

typedef __attribute__((ext_vector_type(2))) float v2f;
typedef __attribute__((ext_vector_type(8))) float v8f;

#define DF 32  // feature dim == wave32 width

// ---------------- utility fills (graph-capture safe, deterministic) ---------
__global__ void fill_kernel(float* __restrict__ p, float v, int n) {
  int i = blockIdx.x * blockDim.x + threadIdx.x;
  if (i < n) p[i] = v;
}

// ---------------- degree (by dst, +self-loop via init=1) and count (by src) -
__global__ __launch_bounds__(256) void degree_count_kernel(
    const int* __restrict__ src, const int* __restrict__ dst,
    float* __restrict__ deg,   // pre-filled with 1.0 (self loop)
    float* __restrict__ cnt,   // pre-filled with 0.0
    int e_count) {
  int e = blockIdx.x * blockDim.x + threadIdx.x;
  if (e >= e_count) return;
  atomicAdd(&deg[dst[e]], 1.0f);
  atomicAdd(&cnt[src[e]], 1.0f);
}

__global__ void rsqrt_inplace_kernel(float* __restrict__ d, int n) {
  int i = blockIdx.x * blockDim.x + threadIdx.x;
  if (i < n) d[i] = rsqrtf(d[i]);  // deg >= 1 guaranteed by init
}

// ---------------- H = X @ W via V_WMMA_F32_16X16X4_F32 (exact f32) ----------
// One wave per 16-row M-tile; K=32 -> 8 chained WMMAs; two 16-col N-tiles.
// A 16x4 f32 layout (ISA 7.12.2): lanes 0-15 hold K=0,1; lanes 16-31 K=2,3.
// B 4x16 layout: VGPR0 lanes0-15 K-row0, lanes16-31 K-row2; VGPR1 rows 1,3.
// C/D 16x16 f32: VGPR v, lanes0-15 -> M=v, lanes16-31 -> M=v+8.
__global__ __launch_bounds__(256) void gemm_wmma_kernel(
    const float* __restrict__ X, const float* __restrict__ W,
    float* __restrict__ H, int n_rows) {
  const int lane = threadIdx.x & 31;
  const int wave = (blockIdx.x * blockDim.x + threadIdx.x) >> 5;
  const int mtiles = (n_rows + 15) >> 4;
  if (wave >= mtiles) return;             // wave-uniform: EXEC stays all-ones

  const int rowBase = wave << 4;
  const bool hi = lane >= 16;
  int m = rowBase + (lane & 15);
  if (m >= n_rows) m = n_rows - 1;        // clamp loads; stores guarded below
  const int col = lane & 15;

  v8f acc0 = {};
  v8f acc1 = {};
#pragma unroll
  for (int k = 0; k < 8; ++k) {
    const int kb = 4 * k;
    // A fragment: this lane's row m, K = kb + (hi?2:0) .. +1
    const float* ap = X + (size_t)m * DF + kb + (hi ? 2 : 0);
    v2f a;
    a.x = ap[0];
    a.y = ap[1];
    // B fragments for N-tile 0 and 1
    const int krow = kb + (hi ? 2 : 0);
    v2f b0, b1;
    b0.x = W[(size_t)krow * DF + col];
    b0.y = W[(size_t)(krow + 1) * DF + col];
    b1.x = W[(size_t)krow * DF + 16 + col];
    b1.y = W[(size_t)(krow + 1) * DF + 16 + col];
    acc0 = __builtin_amdgcn_wmma_f32_16x16x4_f32(false, a, false, b0,
                                                 (short)0, acc0, false, false);
    acc1 = __builtin_amdgcn_wmma_f32_16x16x4_f32(false, a, false, b1,
                                                 (short)0, acc1, false, false);
  }
#pragma unroll
  for (int v = 0; v < 8; ++v) {
    const int row = rowBase + v + (hi ? 8 : 0);
    if (row < n_rows) {
      H[(size_t)row * DF + col]      = acc0[v];
      H[(size_t)row * DF + 16 + col] = acc1[v];
    }
  }
}

// ---------------- edge-parallel GCN aggregation (wave per edge, lane=feat) --
__global__ __launch_bounds__(256) void aggregate_kernel(
    const int* __restrict__ src, const int* __restrict__ dst,
    const float* __restrict__ H, const float* __restrict__ dinv,
    float* __restrict__ agg, int e_count) {
  const int e = blockIdx.x * 8 + (threadIdx.x >> 5);
  if (e >= e_count) return;               // wave-uniform
  const int d = threadIdx.x & 31;
  const int s = src[e];
  const int t = dst[e];
  const float v = H[(size_t)s * DF + d] * dinv[s];   // dinv[dst] factored out
  atomicAdd(&agg[(size_t)t * DF + d], v);            // no-return f32 atomic
}

// ---------------- hr = relu(agg*dinv + H*dinv^2 + b)  (in place over agg) ---
__global__ void finalize_relu_kernel(float* __restrict__ agg,
                                     const float* __restrict__ H,
                                     const float* __restrict__ dinv,
                                     const float* __restrict__ b, int total) {
  int i = blockIdx.x * blockDim.x + threadIdx.x;
  if (i >= total) return;
  const int n = i >> 5;
  const int d = i & 31;
  const float di = dinv[n];
  const float v = agg[i] * di + H[i] * (di * di) + b[d];
  agg[i] = fmaxf(v, 0.0f);
}

// ---------------- e = |hr[src]-hr[dst]|^2, scatter-sum by src ---------------
__global__ __launch_bounds__(256) void edge_pow_kernel(
    const int* __restrict__ src, const int* __restrict__ dst,
    const float* __restrict__ hr, float* __restrict__ sbuf, int e_count) {
  const int e = blockIdx.x * 8 + (threadIdx.x >> 5);
  if (e >= e_count) return;
  const int d = threadIdx.x & 31;
  const int s = src[e];
  const int t = dst[e];
  const float diff = hr[(size_t)s * DF + d] - hr[(size_t)t * DF + d];
  atomicAdd(&sbuf[(size_t)s * DF + d], diff * diff);  // |x|^2 == x*x
}

// ---------------- out = tanh(s / max(cnt,1)) --------------------------------
__global__ void mean_tanh_kernel(const float* __restrict__ sbuf,
                                 const float* __restrict__ cnt,
                                 float* __restrict__ out, int total) {
  int i = blockIdx.x * blockDim.x + threadIdx.x;
  if (i >= total) return;
  const int n = i >> 5;
  const float c = fmaxf(cnt[n], 1.0f);
  out[i] = tanhf(sbuf[i] / c);
}

extern "C" void kernel_launch(void* const* d_in, const int* in_sizes, int n_in,
                              void* d_out, int out_size, void* d_ws, size_t ws_size,
                              hipStream_t stream) {
  const float* X  = (const float*)d_in[0];
  const int*   ei = (const int*)d_in[1];
  const float* W  = (const float*)d_in[2];
  const float* b  = (const float*)d_in[3];
  float* out = (float*)d_out;

  const int n = in_sizes[0] / DF;   // 50000
  const int e = in_sizes[1] / 2;    // 1600000
  const int nd = n * DF;

  const int* src = ei;
  const int* dst = ei + e;

  // workspace layout (floats): H | agg(->hr) | s | dinv | cnt
  float* ws   = (float*)d_ws;
  float* H    = ws;
  float* agg  = H + (size_t)nd;
  float* sbuf = agg + (size_t)nd;
  float* dinv = sbuf + (size_t)nd;
  float* cnt  = dinv + n;

  const int B = 256;
  const int gnd = (nd + B - 1) / B;
  const int gn  = (n + B - 1) / B;
  const int ge  = (e + B - 1) / B;
  const int gew = (e + 7) / 8;                    // wave-per-edge grids
  const int gmm = ((((n + 15) / 16) * 32) + B - 1) / B;

  // init
  fill_kernel<<<gnd, B, 0, stream>>>(agg, 0.0f, nd);
  fill_kernel<<<gnd, B, 0, stream>>>(sbuf, 0.0f, nd);
  fill_kernel<<<gn, B, 0, stream>>>(dinv, 1.0f, n);   // self-loop degree
  fill_kernel<<<gn, B, 0, stream>>>(cnt, 0.0f, n);

  // degrees / counts, then dinv = rsqrt(deg) in place
  degree_count_kernel<<<ge, B, 0, stream>>>(src, dst, dinv, cnt, e);
  rsqrt_inplace_kernel<<<gn, B, 0, stream>>>(dinv, n);

  // H = X @ W  (WMMA f32)
  gemm_wmma_kernel<<<gmm, B, 0, stream>>>(X, W, H, n);

  // GCN aggregate + finalize (relu) -> hr stored in agg
  aggregate_kernel<<<gew, B, 0, stream>>>(src, dst, H, dinv, agg, e);
  finalize_relu_kernel<<<gnd, B, 0, stream>>>(agg, H, dinv, b, nd);

  // edge |diff|^2 scatter-sum, then mean + tanh
  edge_pow_kernel<<<gew, B, 0, stream>>>(src, dst, agg, sbuf, e);
  mean_tanh_kernel<<<gnd, B, 0, stream>>>(sbuf, cnt, out, nd);
}